// PrRoIPool_35802847379875
// MI455X (gfx1250) — compile-verified
//
#include <hip/hip_runtime.h>

typedef float v2f __attribute__((ext_vector_type(2)));
typedef float v8f __attribute__((ext_vector_type(8)));

#define OUT_HW 7
#define SCALE 0.25f
#define CC 64
#define HH 100
#define WW 100
#define KPAD 128
#define WSTRIDE 130   // padded weight-row stride: 8B-aligned float2, bank-spread

__device__ __forceinline__ float hat_cdf(float t) {
    float a = fminf(fmaxf(t + 1.0f, 0.0f), 1.0f);
    float b = fminf(fmaxf(1.0f - t, 0.0f), 1.0f);
    return (t < 0.0f) ? (0.5f * a * a) : (1.0f - 0.5f * b * b);
}

__global__ __launch_bounds__(128)
void prroi_wmma_f32_kernel(const float* __restrict__ feat,
                           const float* __restrict__ rois,
                           float* __restrict__ out,
                           int batch)
{
    __shared__ float sWx[8][WSTRIDE];          // [q][w]  (row 7 zero)
    __shared__ float sWy[8][WSTRIDE];          // [p][h]  (row 7 zero)
    __shared__ float sU[4][KPAD][16];          // per-wave intermediate U[h][q]

    const int n    = blockIdx.x;
    const int tid  = threadIdx.x;
    const int lane = tid & 31;
    const int wave = tid >> 5;
    const int m16  = lane & 15;
    const int hi16 = (lane >> 4) & 1;

    // ---- ROI parameters (uniform per block) ----
    const float* r = rois + (size_t)n * 5;
    int bi = (int)r[0];
    bi = max(0, min(batch - 1, bi));
    float x1 = r[1] * SCALE, y1 = r[2] * SCALE;
    float x2 = r[3] * SCALE, y2 = r[4] * SCALE;
    float rw = fmaxf(x2 - x1, 0.0f), rh = fmaxf(y2 - y1, 0.0f);
    float bw = rw / OUT_HW, bh = rh / OUT_HW;
    float area = bw * bh;
    float sc = (area > 0.0f) ? (1.0f / fmaxf(area, 1e-12f)) : 0.0f;

    // ---- exact hat-kernel integral weights into LDS ----
    for (int idx = tid; idx < 2 * 8 * WSTRIDE; idx += blockDim.x) {
        int which = idx / (8 * WSTRIDE);             // 0 = Wx, 1 = Wy
        int rem   = idx - which * 8 * WSTRIDE;
        int q     = rem / WSTRIDE;
        int k     = rem - q * WSTRIDE;
        float val = 0.0f;
        if (q < OUT_HW && k < (which ? HH : WW)) {
            float start = which ? y1 : x1;
            float bs    = which ? bh : bw;
            float u0 = start + (float)q * bs;
            float u1 = u0 + bs;
            float px = (float)k;
            val = hat_cdf(u1 - px) - hat_cdf(u0 - px);
        }
        if (which) sWy[q][k] = val; else sWx[q][k] = val;
    }
    __syncthreads();

    // zero U rows 112..127 once (never rewritten; stage-2 K covers 0..127)
    for (int idx = lane; idx < 16 * 16; idx += 32)
        sU[wave][112 + (idx >> 4)][idx & 15] = 0.0f;

    // ---- hoist Wx fragments (stage-1 B, reused 7 tiles x 8 channels) ----
    v2f bx[32];
    #pragma unroll
    for (int ks = 0; ks < 32; ++ks) {
        int k0 = ks * 4 + hi16 * 2;                  // 32-bit frag: lane-half -> K+2
        if (m16 < 8) { bx[ks].x = sWx[m16][k0]; bx[ks].y = sWx[m16][k0 + 1]; }
        else         { bx[ks] = (v2f){0.0f, 0.0f}; }
    }

    const float* fbase = feat + (size_t)bi * CC * (HH * WW);

    for (int i = 0; i < 8; ++i) {
        int c = blockIdx.y * 32 + wave * 8 + i;
        const float* fc = fbase + (size_t)c * (HH * WW);

        // ---- stage 1: U[h][q] = sum_w f[h][w] * Wx[q][w] ----
        for (int ht = 0; ht < 7; ++ht) {             // h tiles cover 0..111
            int h = ht * 16 + m16;                   // A-matrix row (both lane halves)
            const float* frow = fc + h * WW;
            bool hok = (h < HH);

            // batch all A-fragment loads first so their latency overlaps
            v2f av[32];
            #pragma unroll
            for (int ks = 0; ks < 32; ++ks) {
                int k0 = ks * 4 + hi16 * 2;          // even -> 8B-aligned float2
                if (hok && k0 < WW) { av[ks].x = frow[k0]; av[ks].y = frow[k0 + 1]; }
                else                { av[ks] = (v2f){0.0f, 0.0f}; }
            }
            // two independent accumulation chains
            v8f acc0 = {}, acc1 = {};
            #pragma unroll
            for (int ks = 0; ks < 32; ks += 2) {
                acc0 = __builtin_amdgcn_wmma_f32_16x16x4_f32(
                         false, av[ks],     false, bx[ks],     (short)0, acc0, false, false);
                acc1 = __builtin_amdgcn_wmma_f32_16x16x4_f32(
                         false, av[ks + 1], false, bx[ks + 1], (short)0, acc1, false, false);
            }
            // D layout: lane holds M = r + hi16*8, N = lane%16
            #pragma unroll
            for (int rr = 0; rr < 8; ++rr)
                sU[wave][ht * 16 + rr + hi16 * 8][m16] = acc0[rr] + acc1[rr];
        }

        __builtin_amdgcn_sched_barrier(0);           // keep DS store->load order

        // ---- stage 2: out[p][q] = sum_h Wy[p][h] * U[h][q] ----
        // batch LDS reads (Wy A-frags + U B-frags), then run WMMA chains
        v2f ay[32], bu[32];
        #pragma unroll
        for (int ks = 0; ks < 32; ++ks) {
            int k0 = ks * 4 + hi16 * 2;
            if (m16 < 8) { ay[ks].x = sWy[m16][k0]; ay[ks].y = sWy[m16][k0 + 1]; }
            else         { ay[ks] = (v2f){0.0f, 0.0f}; }
            bu[ks].x = sU[wave][k0][m16];
            bu[ks].y = sU[wave][k0 + 1][m16];
        }
        v8f acc2 = {}, acc3 = {};
        #pragma unroll
        for (int ks = 0; ks < 32; ks += 2) {
            acc2 = __builtin_amdgcn_wmma_f32_16x16x4_f32(
                     false, ay[ks],     false, bu[ks],     (short)0, acc2, false, false);
            acc3 = __builtin_amdgcn_wmma_f32_16x16x4_f32(
                     false, ay[ks + 1], false, bu[ks + 1], (short)0, acc3, false, false);
        }
        __builtin_amdgcn_sched_barrier(0);

        // ---- scaled store of the valid 7x7 corner ----
        #pragma unroll
        for (int rr = 0; rr < 8; ++rr) {
            int p = rr + hi16 * 8;
            int q = m16;
            if (p < OUT_HW && q < OUT_HW)
                out[(((size_t)n * CC + c) * OUT_HW + p) * OUT_HW + q] =
                    (acc2[rr] + acc3[rr]) * sc;
        }
    }
}

extern "C" void kernel_launch(void* const* d_in, const int* in_sizes, int n_in,
                              void* d_out, int out_size, void* d_ws, size_t ws_size,
                              hipStream_t stream) {
    const float* feat = (const float*)d_in[0];
    const float* rois = (const float*)d_in[1];
    float* out = (float*)d_out;
    int n_rois = in_sizes[1] / 5;
    int batch  = in_sizes[0] / (CC * HH * WW);
    dim3 grid(n_rois, 2);
    prroi_wmma_f32_kernel<<<grid, 128, 0, stream>>>(feat, rois, out, batch);
}